// NBP_2_oc_40226663694715
// MI455X (gfx1250) — compile-verified
//
#include <hip/hip_runtime.h>
#include <math.h>

#define NN   882
#define N2   1764
#define MOC  1024
#define BB   8
#define TT   6
#define KG   24
#define CAP  32          // max nnz per H row (mean ~6; P(>32) ~ 1e-17)
#define CLIPV 30.0f

typedef float v2f __attribute__((ext_vector_type(2)));
typedef float v4f __attribute__((ext_vector_type(4)));
typedef float v8f __attribute__((ext_vector_type(8)));

__device__ __forceinline__ float sgn1(float x) {
    // reference: where(x==0, 1, sign(x))
    return (x == 0.0f) ? 1.0f : (x > 0.0f ? 1.0f : -1.0f);
}

// ---------------------------------------------------------------------------
// Build CSR of H (1024 x 1764, ~6 nnz/row). One thread per row.
// ---------------------------------------------------------------------------
__global__ void k_build_csr(const float* __restrict__ H,
                            int* __restrict__ rowcnt, int* __restrict__ rowcols) {
    int i = blockIdx.x * blockDim.x + threadIdx.x;
    if (i >= MOC) return;
    const float* row = H + (size_t)i * N2;
    int c = 0;
    for (int j = 0; j < N2; ++j) {
        if (row[j] != 0.0f) { if (c < CAP) rowcols[i * CAP + c] = j; ++c; }
    }
    rowcnt[i] = c < CAP ? c : CAP;
}

// ---------------------------------------------------------------------------
// Syndrome: sgn_syn[b,i] = 1 - 2*mod2( rows 0..511: errorz@Hx^T ; 512..: errorx@Hz^T )
// WMMA f32 16x16x4. Main K loop unconditional (float2 b64 loads), masked tail.
// ---------------------------------------------------------------------------
__global__ void k_syndrome(const float* __restrict__ errorx,
                           const float* __restrict__ errorz,
                           const float* __restrict__ Hx,
                           const float* __restrict__ Hz,
                           float* __restrict__ sgn_syn) {
    int tile = blockIdx.x;              // 64 tiles of 16 check rows
    int i0   = tile * 16;
    const float* E; const float* Hmat; int ir0 = i0;
    if (i0 < 512) { E = errorz; Hmat = Hx; }
    else          { E = errorx; Hmat = Hz; ir0 = i0 - 512; }
    int lane = threadIdx.x & 31;
    int mn   = lane & 15;               // A: M index, B: N index
    int kh   = (lane >> 4) << 1;        // K base within quad: 0 or 2
    float am = (mn < BB) ? 1.0f : 0.0f; // mask padded batch rows
    const float* Erow = E + (size_t)(mn < BB ? mn : 0) * NN;   // safe pointer
    const float* Hrow = Hmat + (size_t)(ir0 + mn) * NN;
    v8f acc = {};
    for (int k0 = 0; k0 < 880; k0 += 4) {              // ka<=878, kb<=879 in range
        int ka = k0 + kh;
        v2f a = *(const v2f*)(Erow + ka);
        v2f b = *(const v2f*)(Hrow + ka);
        a.x *= am; a.y *= am;
        acc = __builtin_amdgcn_wmma_f32_16x16x4_f32(false, a, false, b,
                                                    (short)0, acc, false, false);
    }
    {   // tail k0 = 880 : K = 880..883, mask K>=882
        int ka = 880 + kh, kb = ka + 1;
        int kaS = ka < NN ? ka : 0, kbS = kb < NN ? kb : 0;
        float m0 = (ka < NN) ? am : 0.0f, m1 = (kb < NN) ? am : 0.0f;
        v2f a, b;
        a.x = Erow[kaS] * m0;                a.y = Erow[kbS] * m1;
        b.x = (ka < NN) ? Hrow[kaS] : 0.0f;  b.y = (kb < NN) ? Hrow[kbS] : 0.0f;
        acc = __builtin_amdgcn_wmma_f32_16x16x4_f32(false, a, false, b,
                                                    (short)0, acc, false, false);
    }
    // D layout: lanes 0..15 -> N=lane, VGPR v -> M=v  => batch b=v, row i0+lane
    if (lane < 16) {
        int irow = i0 + lane;
        #pragma unroll
        for (int v = 0; v < BB; ++v) {
            float s = acc[v];
            float r = s - 2.0f * floorf(s * 0.5f);   // exact mod-2 of integer float
            sgn_syn[v * MOC + irow] = 1.0f - 2.0f * r;
        }
    }
}

// ---------------------------------------------------------------------------
// Init: c2v_0 = 0, Gamma_0 = llr * w_llr[0]
// ---------------------------------------------------------------------------
__global__ void k_init_state(const float* __restrict__ ep,
                             const float* __restrict__ w_llr,
                             float* __restrict__ c2vA, float* __restrict__ gammaA) {
    int idx = blockIdx.x * blockDim.x + threadIdx.x;
    if (idx < BB * MOC * CAP) c2vA[idx] = 0.0f;
    if (idx < BB * N2) {
        float e = ep[0];
        gammaA[idx] = logf((1.0f - e) / e) * w_llr[0];
    }
}

__global__ void k_init_gamma(const float* __restrict__ ep,
                             const float* __restrict__ w_llr, int t1,
                             float* __restrict__ gammaN) {
    int idx = blockIdx.x * blockDim.x + threadIdx.x;
    if (idx < BB * N2) {
        float e = ep[0];
        gammaN[idx] = logf((1.0f - e) / e) * w_llr[t1];
    }
}

// ---------------------------------------------------------------------------
// CN update + Gamma accumulation for one BP iteration.
// Grid: MOC blocks (one per check row i), 256 threads = 8 waves (wave b = batch).
// w_vn row staged into LDS with gfx1250 async global->LDS b128 (ASYNCcnt).
// Dense sign product factorized: base over all j (c2vPrev==0 off support),
// then fix the <=nnz support entries (all factors +-1 so fixup = re-multiply).
// ---------------------------------------------------------------------------
__global__ void __launch_bounds__(256) k_cn(
        const float* __restrict__ gammaPrev, const float* __restrict__ c2vPrev,
        const float* __restrict__ w_vn_t,    const float* __restrict__ w_cn_t,
        const float* __restrict__ sgn_syn,
        const int*   __restrict__ rowcnt,    const int* __restrict__ rowcols,
        float* __restrict__ c2vNew,          float* __restrict__ gammaNext) {
    __shared__ __align__(16) float s_w[N2];
    __shared__ int s_cols[CAP];
    __shared__ int s_nnz;
    const int i   = blockIdx.x;
    const int tid = threadIdx.x;

    // --- async global->LDS copy of w_vn[t] row i (7056 B = 441 x b128) ---
    {
        unsigned long long gbase = (unsigned long long)(uintptr_t)(w_vn_t + (size_t)i * N2);
        unsigned lds0 = (unsigned)(uintptr_t)&s_w[0];   // low 32 bits = LDS byte offset
        for (int c = tid; c < (N2 * 4) / 16; c += 256) {
            unsigned ldsa = lds0 + (unsigned)(c * 16);
            unsigned voff = (unsigned)(c * 16);
            asm volatile("global_load_async_to_lds_b128 %0, %1, %2"
                         :: "v"(ldsa), "v"(voff), "s"(gbase) : "memory");
        }
        asm volatile("s_wait_asynccnt 0x0" ::: "memory");
    }
    if (tid < CAP) s_cols[tid] = rowcols[i * CAP + tid];
    if (tid == 0)  s_nnz = rowcnt[i];
    __syncthreads();

    const int b    = tid >> 5;
    const int lane = tid & 31;
    const float* G = gammaPrev + (size_t)b * N2;
    __builtin_prefetch(c2vPrev + ((size_t)b * MOC + i) * CAP, 0, 3);

    // Base sign product over all 1764 columns, float4-vectorized (441 quads).
    float p = 1.0f;
    for (int q = lane; q < N2 / 4; q += 32) {
        v4f g = *(const v4f*)(G + q * 4);
        v4f w = *(const v4f*)(&s_w[q * 4]);
        p *= sgn1(g.x * w.x) * sgn1(g.y * w.y) * sgn1(g.z * w.z) * sgn1(g.w * w.w);
    }
    #pragma unroll
    for (int off = 16; off; off >>= 1) p *= __shfl_xor(p, off, 32);

    const int nnz  = s_nnz;
    const bool act = lane < nnz;
    int   j    = act ? s_cols[lane] : 0;
    float c2vp = act ? c2vPrev[((size_t)b * MOC + i) * CAP + lane] : 0.0f;
    float g    = act ? G[j] : 0.0f;
    float wv   = act ? s_w[j] : 0.0f;
    float v    = fminf(fmaxf(g - c2vp, -CLIPV), CLIPV) * wv;
    float sgn  = act ? sgn1(v) : 1.0f;               // zeros -> +1 (matches ref)
    float phi  = (act && v != 0.0f) ? -logf(tanhf(0.5f * fabsf(v))) : 0.0f;
    float fix  = act ? sgn1(g * wv) * sgn : 1.0f;    // cancel base factor, apply true one

    float asum = phi, fp = fix;
    #pragma unroll
    for (int off = 16; off; off >>= 1) {
        asum += __shfl_xor(asum, off, 32);
        fp   *= __shfl_xor(fp,   off, 32);
    }
    const float sp = p * fp;                          // full-row sign product

    if (act) {
        float first  = sp * sgn;                      // prod(sign)/sign for +-1
        float s      = asum - phi;                    // sum of other phi (>=0)
        float second = (s == 0.0f) ? 0.0f : -logf(tanhf(0.5f * s));
        float out    = first * second * sgn_syn[b * MOC + i] * w_cn_t[i];
        c2vNew[((size_t)b * MOC + i) * CAP + lane] = out;
        atomicAdd(&gammaNext[(size_t)b * N2 + j], out);
    }
}

// ---------------------------------------------------------------------------
// iter_loss: corr = where(error==1, sigmoid(-Gamma_part), 0); syn_real = corr@G^T
// loss[b] = sum_n |sin(pi/2 * syn_real)|. 4 waves: {Gx,Gz} x {n0=0,16}.
// Main K loop unconditional float2 loads; masked tail.
// ---------------------------------------------------------------------------
__global__ void __launch_bounds__(128) k_loss(
        const float* __restrict__ gamma,  const float* __restrict__ errorx,
        const float* __restrict__ errorz, const float* __restrict__ Gx,
        const float* __restrict__ Gz,     float* __restrict__ loss_t) {
    __shared__ float s_loss[BB];
    int tid = threadIdx.x;
    if (tid < BB) s_loss[tid] = 0.0f;
    __syncthreads();
    int w    = tid >> 5;
    int lane = tid & 31;
    int part = w >> 1;                 // 0: corrz=errorx&pX @Gx^T ; 1: corrx=errorz&pZ @Gz^T
    int n0   = (w & 1) * 16;
    const float* err = part ? errorz : errorx;
    const float* Gm  = part ? Gz : Gx;
    int goff = part * NN;
    int mn   = lane & 15;
    int kh   = (lane >> 4) << 1;
    int ng   = n0 + mn;
    float am = (mn < BB) ? 1.0f : 0.0f;
    float bm = (ng < KG) ? 1.0f : 0.0f;
    const float* errRow = err   + (size_t)(mn < BB ? mn : 0) * NN;
    const float* gRow   = gamma + (size_t)(mn < BB ? mn : 0) * N2 + goff;
    const float* Grow   = Gm    + (size_t)(ng < KG ? ng : 0) * NN;
    v8f acc = {};
    for (int k0 = 0; k0 < 880; k0 += 4) {
        int ka = k0 + kh;
        v2f e  = *(const v2f*)(errRow + ka);
        v2f gv = *(const v2f*)(gRow + ka);
        v2f gw = *(const v2f*)(Grow + ka);
        v2f a, b;
        a.x = (e.x == 1.0f ? am : 0.0f) / (1.0f + expf(gv.x));
        a.y = (e.y == 1.0f ? am : 0.0f) / (1.0f + expf(gv.y));
        b.x = gw.x * bm;  b.y = gw.y * bm;
        acc = __builtin_amdgcn_wmma_f32_16x16x4_f32(false, a, false, b,
                                                    (short)0, acc, false, false);
    }
    {   // tail k0 = 880
        int ka = 880 + kh, kb = ka + 1;
        int kaS = ka < NN ? ka : 0, kbS = kb < NN ? kb : 0;
        float m0 = (ka < NN) ? 1.0f : 0.0f, m1 = (kb < NN) ? 1.0f : 0.0f;
        v2f a, b;
        a.x = (errRow[kaS] == 1.0f ? am * m0 : 0.0f) / (1.0f + expf(gRow[kaS]));
        a.y = (errRow[kbS] == 1.0f ? am * m1 : 0.0f) / (1.0f + expf(gRow[kbS]));
        b.x = Grow[kaS] * bm * m0;  b.y = Grow[kbS] * bm * m1;
        acc = __builtin_amdgcn_wmma_f32_16x16x4_f32(false, a, false, b,
                                                    (short)0, acc, false, false);
    }
    if (lane < 16 && (n0 + lane) < KG) {
        #pragma unroll
        for (int v = 0; v < BB; ++v)
            atomicAdd(&s_loss[v], fabsf(sinf(1.57079632679f * acc[v])));
    }
    __syncthreads();
    if (tid < BB) loss_t[tid] = s_loss[tid];
}

// ---------------------------------------------------------------------------
// argmin over T, cumsum/(idx+1), mean over batch -> scalar
// ---------------------------------------------------------------------------
__global__ void k_finalize(const float* __restrict__ loss_arr, float* __restrict__ out) {
    int lane = threadIdx.x;
    float per = 0.0f;
    if (lane < BB) {
        float runsum = 0.0f, best = 3.4e38f, bestCsum = 0.0f; int bestIdx = 0;
        for (int t = 0; t < TT; ++t) {
            float v = loss_arr[t * BB + lane];
            runsum += v;
            if (v < best) { best = v; bestIdx = t; bestCsum = runsum; }  // first-min
        }
        per = bestCsum / (float)(bestIdx + 1);
    }
    #pragma unroll
    for (int off = 16; off; off >>= 1) per += __shfl_xor(per, off, 32);
    if (lane == 0) out[0] = per / (float)BB;
}

// ---------------------------------------------------------------------------
extern "C" void kernel_launch(void* const* d_in, const int* in_sizes, int n_in,
                              void* d_out, int out_size, void* d_ws, size_t ws_size,
                              hipStream_t stream) {
    (void)in_sizes; (void)n_in; (void)out_size; (void)ws_size;
    const float* errorx = (const float*)d_in[0];
    const float* errorz = (const float*)d_in[1];
    const float* ep     = (const float*)d_in[2];
    const float* H      = (const float*)d_in[3];
    const float* Hx     = (const float*)d_in[4];
    const float* Hz     = (const float*)d_in[5];
    const float* Gx     = (const float*)d_in[6];
    const float* Gz     = (const float*)d_in[7];
    const float* w_vn   = (const float*)d_in[8];
    const float* w_llr  = (const float*)d_in[9];
    const float* w_cn   = (const float*)d_in[10];

    char* ws = (char*)d_ws;
    size_t off = 0;
    auto alloc = [&](size_t bytes) {
        size_t start = (off + 255) & ~(size_t)255;
        off = start + bytes;
        return (void*)(ws + start);
    };
    int*   rowcnt  = (int*)  alloc(MOC * sizeof(int));
    int*   rowcols = (int*)  alloc((size_t)MOC * CAP * sizeof(int));
    float* sgnsyn  = (float*)alloc((size_t)BB * MOC * sizeof(float));
    float* gammaA  = (float*)alloc((size_t)BB * N2 * sizeof(float));
    float* gammaB  = (float*)alloc((size_t)BB * N2 * sizeof(float));
    float* c2vA    = (float*)alloc((size_t)BB * MOC * CAP * sizeof(float));
    float* c2vB    = (float*)alloc((size_t)BB * MOC * CAP * sizeof(float));
    float* lossArr = (float*)alloc((size_t)TT * BB * sizeof(float));

    k_build_csr<<<(MOC + 255) / 256, 256, 0, stream>>>(H, rowcnt, rowcols);
    k_syndrome<<<MOC / 16, 32, 0, stream>>>(errorx, errorz, Hx, Hz, sgnsyn);
    k_init_state<<<(BB * MOC * CAP + 255) / 256, 256, 0, stream>>>(ep, w_llr, c2vA, gammaA);

    float* gp = gammaA; float* gn = gammaB;
    float* cp = c2vA;   float* cnew = c2vB;
    for (int t = 0; t < TT; ++t) {
        k_init_gamma<<<(BB * N2 + 255) / 256, 256, 0, stream>>>(ep, w_llr, t + 1, gn);
        k_cn<<<MOC, 256, 0, stream>>>(gp, cp,
                                      w_vn + (size_t)t * MOC * N2,
                                      w_cn + (size_t)t * MOC,
                                      sgnsyn, rowcnt, rowcols, cnew, gn);
        k_loss<<<1, 128, 0, stream>>>(gn, errorx, errorz, Gx, Gz, lossArr + t * BB);
        float* tg = gp; gp = gn; gn = tg;
        float* tc = cp; cp = cnew; cnew = tc;
    }
    k_finalize<<<1, 32, 0, stream>>>(lossArr, (float*)d_out);
}